// ChebLayer_90331752169519
// MI455X (gfx1250) — compile-verified
//
#include <hip/hip_runtime.h>

typedef __attribute__((ext_vector_type(2))) float v2f;
typedef __attribute__((ext_vector_type(8))) float v8f;
typedef unsigned int u32x4 __attribute__((ext_vector_type(4)));
typedef int i32x4 __attribute__((ext_vector_type(4)));
typedef int i32x8 __attribute__((ext_vector_type(8)));

#define Bb 128
#define Nn 512
#define Ff 32

// B-operand buffers use a pair-interleaved layout so a WMMA B-fragment
// (rows k,k+1 at column n) is one aligned 8-byte load:
//   packed[(k>>1)*64 + n*2 + (k&1)] = val[k][n]
__device__ __forceinline__ size_t bpack_idx(int row, int f) {
  return (size_t)(row >> 1) * 64 + (size_t)f * 2 + (row & 1);
}

// ---------------- kernel 1: D^{-1/2} and pre-scaled packed x ----------------
// one wave32 per adjacency row; butterfly reduction leaves rowsum in all
// lanes, so the wave also writes xpack = dinv[row]*x[row][f] (packed).
__global__ __launch_bounds__(128) void cheb_dinv_kernel(
    const float* __restrict__ adj, const float* __restrict__ x,
    float* __restrict__ dinv, float* __restrict__ xpack) {
  const int lane = threadIdx.x & 31;
  const int wave = threadIdx.x >> 5;
  const int row  = blockIdx.x * 4 + wave;   // row in [0, B*N)
  if (row >= Bb * Nn) return;
  const float* arow = adj + (size_t)row * Nn;
  float s = 0.f;
  #pragma unroll 4
  for (int j = lane; j < Nn; j += 32) s += arow[j];
  #pragma unroll
  for (int off = 16; off > 0; off >>= 1) s += __shfl_xor(s, off, 32);
  const float dv = (s > 0.f) ? rsqrtf(s) : 0.f;
  if (lane == 0) dinv[row] = dv;
  const int brow = row & (Nn - 1);          // batch-local row
  const size_t base = (size_t)(row >> 9) * Nn * Ff;   // batch offset
  xpack[base + bpack_idx(brow, lane)] = dv * x[(size_t)row * Ff + lane];
}

// ---------------- kernel 2/3: adj-GEMM via TDM + WMMA f32 16x16x4 ------
// A operand = raw adj tile (16 rows x 128 K per wave), DMA'd by the
// Tensor Data Mover into LDS with padded stride 132 dwords (1-dword pad
// every 32) for conflict-free column reads.  B operand = dinv-prescaled
// pair-packed source (xpack or T1sp).  Row scaling in the epilogue:
//   PHASE 0: T1  = -dinv_i * (A @ xpack) ; also T1sp = dinv*T1 (packed)
//   PHASE 1: T2  = -2*dinv_i * (A @ T1sp) - x ;
//            out = relu(x@W0 + T1@W1 + T2@W2 + b) + x
template <int PHASE>
__global__ __launch_bounds__(128) void cheb_gemm_kernel(
    const float* __restrict__ adj, const float* __restrict__ dinv,
    const float* __restrict__ x,   const float* __restrict__ T1,
    const float* __restrict__ Bsrc,                   // xpack or T1sp
    const float* __restrict__ W,   const float* __restrict__ bias,
    float* __restrict__ t1out, float* __restrict__ t1spout,
    float* __restrict__ out) {
  __shared__ float adjraw[4][16 * 132];  // per-wave TDM destination (padded)
  __shared__ float acc[16][32];
  __shared__ float xs[16][32];
  __shared__ float t1s[16][32];
  __shared__ float Wl[3][32][32];
  __shared__ float bl[32];

  const int b    = blockIdx.x >> 5;          // 32 row tiles per batch
  const int i0   = (blockIdx.x & 31) << 4;
  const int tid  = threadIdx.x;
  const int lane = tid & 31;
  const int w    = tid >> 5;
  const int kbase = w * 128;                 // this wave's K range

  const float* adjb = adj + (size_t)b * Nn * Nn;
  const float* dvb  = dinv + (size_t)b * Nn;
  const float* Xb   = Bsrc + (size_t)b * Nn * Ff;   // packed per batch

  // ---- issue TDM: 16x128 f32 tile of adj -> LDS, padded 32+1 dwords ----
  {
    const unsigned lds_off =
        (unsigned)(size_t)(const void*)&adjraw[w][0];       // LDS byte addr
    const unsigned long long ga =
        (unsigned long long)(size_t)(adjb + (size_t)i0 * Nn + kbase);
    u32x4 g0;
    g0[0] = 1u;                                  // count=1, user mode
    g0[1] = lds_off;                             // lds_addr
    g0[2] = (unsigned)(ga & 0xFFFFFFFFull);      // global_addr[31:0]
    g0[3] = (unsigned)((ga >> 32) & 0x1FFFFFFull) | 0x80000000u;  // +type=2
    i32x8 g1;
    g1[0] = 0x01120000;          // data_size=4B, pad_en, interval=32dw, amt=1dw
    g1[1] = (int)(128u << 16);   // tensor_dim0 = 128
    g1[2] = (int)(16u << 16);    // tensor_dim1 = 16
    g1[3] = (int)(128u << 16);   // tile_dim0 = 128
    g1[4] = 16;                  // tile_dim1 = 16, tile_dim2 = 0
    g1[5] = Nn;                  // tensor_dim0_stride = 512 (low 32)
    g1[6] = 0;
    g1[7] = 0;
    i32x4 g2; g2[0] = 1; g2[1] = 1; g2[2] = 0; g2[3] = 0;  // dims2/3 benign
    i32x4 g3; g3[0] = 0; g3[1] = 1 << 16; g3[2] = 0; g3[3] = 0;  // dim4=1
    i32x8 g4;                                               // extra operand
    g4[0] = 0; g4[1] = 0; g4[2] = 0; g4[3] = 0;             // (unused) zeros
    g4[4] = 0; g4[5] = 0; g4[6] = 0; g4[7] = 0;
    __builtin_amdgcn_tensor_load_to_lds(g0, g1, g2, g3, g4, 0);
  }

  // ---- other LDS init while the DMA flies ----
  for (int idx = tid; idx < 16 * 32; idx += 128) acc[idx >> 5][idx & 31] = 0.f;
  if (PHASE == 1) {
    for (int idx = tid; idx < 3 * 32 * 32; idx += 128)
      Wl[idx >> 10][(idx >> 5) & 31][idx & 31] = W[idx];
    if (tid < 32) bl[tid] = bias[tid];
  }
  __syncthreads();

  __builtin_amdgcn_s_wait_tensorcnt(0);       // adj tile resident in LDS

  v8f c0 = {0.f, 0.f, 0.f, 0.f, 0.f, 0.f, 0.f, 0.f};
  v8f c1 = c0;
  const float* Aw = &adjraw[w][0];
  const int m = lane & 15;                    // A row / B column
  const int kh = 2 * (lane >> 4);             // K sub-offset per lane half

  #pragma unroll 8
  for (int s = 0; s < 32; ++s) {
    const int kk = s * 4 + kh;                // local K (0..126, even)
    const int p  = kk + (kk >> 5);            // padded LDS column
    v2f a;
    a.x = Aw[132 * m + p];
    a.y = Aw[132 * m + p + 1];
    const int gk = kbase + kk;                // even -> pair-aligned
    const float* xr = Xb + (size_t)(gk >> 1) * 64;
    const v2f b0 = *(const v2f*)(xr + 2 * m);          // rows gk,gk+1 @ col m
    const v2f b1 = *(const v2f*)(xr + 2 * (16 + m));   // .. @ col 16+m
    c0 = __builtin_amdgcn_wmma_f32_16x16x4_f32(false, a, false, b0,
                                               (short)0, c0, false, false);
    c1 = __builtin_amdgcn_wmma_f32_16x16x4_f32(false, a, false, b1,
                                               (short)0, c1, false, false);
  }

  // ---- cross-wave reduction of C tiles ----
  {
    const int mrow = 8 * (lane >> 4);
    #pragma unroll
    for (int r = 0; r < 8; ++r) {
      atomicAdd(&acc[mrow + r][m], c0[r]);
      atomicAdd(&acc[mrow + r][16 + m], c1[r]);
    }
  }
  __syncthreads();

  if (PHASE == 0) {
    float* T1b   = t1out + ((size_t)b * Nn + i0) * Ff;
    float* T1spb = t1spout + (size_t)b * Nn * Ff;     // packed, batch base
    for (int idx = tid; idx < 512; idx += 128) {
      const int r = idx >> 5, f = idx & 31;
      const float dv = dvb[i0 + r];
      const float t1 = -dv * acc[r][f];       // T1 = -dinv_i * (A@xs)
      T1b[idx] = t1;
      T1spb[bpack_idx(i0 + r, f)] = dv * t1;  // phase-1 packed B operand
    }
  } else {
    const float* xb  = x  + ((size_t)b * Nn + i0) * Ff;
    const float* t1b = T1 + ((size_t)b * Nn + i0) * Ff;
    for (int idx = tid; idx < 512; idx += 128) {
      xs[idx >> 5][idx & 31]  = xb[idx];
      t1s[idx >> 5][idx & 31] = t1b[idx];
    }
    __syncthreads();
    float* ob = out + ((size_t)b * Nn + i0) * Ff;
    for (int idx = tid; idx < 512; idx += 128) {
      const int r = idx >> 5, o = idx & 31;
      const float dv2 = -2.f * dvb[i0 + r];
      float s = bl[o];
      #pragma unroll
      for (int f = 0; f < 32; ++f) {
        const float t2 = dv2 * acc[r][f] - xs[r][f];  // T2 = 2*Lhat@T1 - x
        s += xs[r][f] * Wl[0][f][o] + t1s[r][f] * Wl[1][f][o] + t2 * Wl[2][f][o];
      }
      ob[idx] = fmaxf(s, 0.f) + xs[r][o];             // relu + residual
    }
  }
}

extern "C" void kernel_launch(void* const* d_in, const int* in_sizes, int n_in,
                              void* d_out, int out_size, void* d_ws, size_t ws_size,
                              hipStream_t stream) {
  (void)in_sizes; (void)n_in; (void)out_size; (void)ws_size;
  const float* adj  = (const float*)d_in[0];
  const float* x    = (const float*)d_in[1];
  const float* W    = (const float*)d_in[2];
  const float* bias = (const float*)d_in[3];
  float* out  = (float*)d_out;
  float* dinv = (float*)d_ws;                          // B*N        (256 KB)
  float* xpk  = dinv + (size_t)Bb * Nn;                // B*N*F      (8 MB)
  float* T1   = xpk  + (size_t)Bb * Nn * Ff;           // B*N*F      (8 MB)
  float* T1sp = T1   + (size_t)Bb * Nn * Ff;           // B*N*F      (8 MB)

  cheb_dinv_kernel<<<(Bb * Nn) / 4, 128, 0, stream>>>(adj, x, dinv, xpk);
  cheb_gemm_kernel<0><<<Bb * (Nn / 16), 128, 0, stream>>>(
      adj, dinv, x, T1, xpk, W, bias, T1, T1sp, out);
  cheb_gemm_kernel<1><<<Bb * (Nn / 16), 128, 0, stream>>>(
      adj, dinv, x, T1, T1sp, W, bias, T1, T1sp, out);
}